// Time_Module_32280974197365
// MI455X (gfx1250) — compile-verified
//
#include <hip/hip_runtime.h>
#include <hip/hip_bf16.h>

// Problem dims (fixed by the reference)
#define BB 16
#define HH 2048
#define WW 128

typedef __attribute__((ext_vector_type(2))) float v2f;
typedef __attribute__((ext_vector_type(8))) float v8f;

// ---------------------------------------------------------------------------
// Pass 1: row sums  s[b*H + h] = sum_w x[b,h,w]
// ---------------------------------------------------------------------------
__global__ void __launch_bounds__(256) rowsum_kernel(const float* __restrict__ x,
                                                     float* __restrict__ s) {
    int row = blockIdx.x * 256 + threadIdx.x;   // 0 .. B*H-1
    const float4* xr = (const float4*)(x + (size_t)row * WW);
    float acc = 0.f;
#pragma unroll
    for (int i = 0; i < WW / 4; ++i) {
        float4 v = xr[i];
        acc += v.x + v.y + v.z + v.w;
    }
    s[row] = acc;
}

// ---------------------------------------------------------------------------
// Pass 2: energy[b,h,g] = cA*<x_h,x_g> + cB*s_h + cC*s_g + cD
// Gram matrix via V_WMMA_F32_16X16X4_F32. Block tile 128x128, 8 waves,
// wave tile 32(M) x 64(N). Both X tiles staged in LDS (stride 132 pad ->
// bank-conflict-free float2 loads).
// ---------------------------------------------------------------------------
__global__ void __launch_bounds__(256) energy_kernel(
    const float* __restrict__ x, const float* __restrict__ s,
    const float* __restrict__ w1, const float* __restrict__ b1,
    const float* __restrict__ w2, const float* __restrict__ b2,
    float* __restrict__ E) {
    extern __shared__ float lds[];
    float* lA = lds;               // 128 x 132
    float* lB = lds + 128 * 132;   // 128 x 132

    const int gb = blockIdx.x, hb = blockIdx.y, b = blockIdx.z;

    // Scalar coefficients (cheap, every thread)
    float cA = 0.f, cB = 0.f, cC = 0.f, cD = 0.f;
#pragma unroll
    for (int c = 0; c < 8; ++c) {
        cA += w1[c] * w2[c];
        cB += w1[c] * b2[c];
        cC += b1[c] * w2[c];
        cD += b1[c] * b2[c];
    }
    cD *= (float)WW;

    const int t = threadIdx.x;
    // Stage A (rows hb*128..+128) and B (rows gb*128..+128) tiles of x
    {
        int row = t >> 1, half = t & 1;
        const float4* srcA =
            (const float4*)(x + ((size_t)b * HH + hb * 128 + row) * WW + half * 64);
        const float4* srcB =
            (const float4*)(x + ((size_t)b * HH + gb * 128 + row) * WW + half * 64);
        float4* dA = (float4*)(lA + row * 132 + half * 64);
        float4* dB = (float4*)(lB + row * 132 + half * 64);
#pragma unroll
        for (int i = 0; i < 16; ++i) {
            dA[i] = srcA[i];
            dB[i] = srcB[i];
        }
    }
    __syncthreads();

    const int lane = t & 31, wv = t >> 5;
    const int wm = wv & 3;    // 4 M-groups of 32 rows
    const int wn = wv >> 2;   // 2 N-groups of 64 cols
    const int mlo = lane & 15, half2 = lane >> 4;   // A/B: k-base = half2*2

    v8f acc[2][4] = {};

    const float* pA0 = lA + (wm * 32 + mlo) * 132 + half2 * 2;
    const float* pA1 = pA0 + 16 * 132;
    const float* pB0 = lB + (wn * 64 + mlo) * 132 + half2 * 2;

#pragma unroll 8
    for (int kk = 0; kk < WW; kk += 4) {
        v2f a0 = *(const v2f*)(pA0 + kk);
        v2f a1 = *(const v2f*)(pA1 + kk);
        v2f bv0 = *(const v2f*)(pB0 + kk);
        v2f bv1 = *(const v2f*)(pB0 + 16 * 132 + kk);
        v2f bv2 = *(const v2f*)(pB0 + 32 * 132 + kk);
        v2f bv3 = *(const v2f*)(pB0 + 48 * 132 + kk);
        acc[0][0] = __builtin_amdgcn_wmma_f32_16x16x4_f32(false, a0, false, bv0, (short)0, acc[0][0], false, false);
        acc[0][1] = __builtin_amdgcn_wmma_f32_16x16x4_f32(false, a0, false, bv1, (short)0, acc[0][1], false, false);
        acc[0][2] = __builtin_amdgcn_wmma_f32_16x16x4_f32(false, a0, false, bv2, (short)0, acc[0][2], false, false);
        acc[0][3] = __builtin_amdgcn_wmma_f32_16x16x4_f32(false, a0, false, bv3, (short)0, acc[0][3], false, false);
        acc[1][0] = __builtin_amdgcn_wmma_f32_16x16x4_f32(false, a1, false, bv0, (short)0, acc[1][0], false, false);
        acc[1][1] = __builtin_amdgcn_wmma_f32_16x16x4_f32(false, a1, false, bv1, (short)0, acc[1][1], false, false);
        acc[1][2] = __builtin_amdgcn_wmma_f32_16x16x4_f32(false, a1, false, bv2, (short)0, acc[1][2], false, false);
        acc[1][3] = __builtin_amdgcn_wmma_f32_16x16x4_f32(false, a1, false, bv3, (short)0, acc[1][3], false, false);
    }

    // Epilogue: C/D layout -> vgpr r holds M = r + half2*8, N = mlo
    const float* srow = s + (size_t)b * HH;
#pragma unroll
    for (int tm = 0; tm < 2; ++tm) {
        int hbase = hb * 128 + wm * 32 + tm * 16 + half2 * 8;
#pragma unroll
        for (int tn = 0; tn < 4; ++tn) {
            int gcol = gb * 128 + wn * 64 + tn * 16 + mlo;
            float sg = srow[gcol];
#pragma unroll
            for (int r = 0; r < 8; ++r) {
                int h = hbase + r;
                float e = cA * acc[tm][tn][r] + cB * srow[h] + cC * sg + cD;
                E[((size_t)b * HH + h) * HH + gcol] = e;
            }
        }
    }
}

// ---------------------------------------------------------------------------
// Pass 3: per-row min/max + exp-sum.  One 256-thread block per row.
// e' = (e - mn) * inv, inv = 1/(mx - mn + 1e-8);  Z = sum exp(e')  (e' in [0,1])
// ---------------------------------------------------------------------------
__global__ void __launch_bounds__(256) stats_kernel(const float* __restrict__ E,
                                                    float* __restrict__ mn_o,
                                                    float* __restrict__ inv_o,
                                                    float* __restrict__ invZ_o) {
    __shared__ float red[256];
    int row = blockIdx.x;
    int t = threadIdx.x;
    const float* e = E + (size_t)row * HH;
    float v[8];
#pragma unroll
    for (int i = 0; i < 8; ++i) v[i] = e[t + i * 256];

    float mn = v[0], mx = v[0];
#pragma unroll
    for (int i = 1; i < 8; ++i) { mn = fminf(mn, v[i]); mx = fmaxf(mx, v[i]); }

    red[t] = mn; __syncthreads();
    for (int o = 128; o > 0; o >>= 1) { if (t < o) red[t] = fminf(red[t], red[t + o]); __syncthreads(); }
    mn = red[0]; __syncthreads();

    red[t] = mx; __syncthreads();
    for (int o = 128; o > 0; o >>= 1) { if (t < o) red[t] = fmaxf(red[t], red[t + o]); __syncthreads(); }
    mx = red[0]; __syncthreads();

    float inv = 1.0f / (mx - mn + 1e-8f);
    float ssum = 0.f;
#pragma unroll
    for (int i = 0; i < 8; ++i) ssum += __expf((v[i] - mn) * inv);

    red[t] = ssum; __syncthreads();
    for (int o = 128; o > 0; o >>= 1) { if (t < o) red[t] += red[t + o]; __syncthreads(); }

    if (t == 0) {
        mn_o[row] = mn;
        inv_o[row] = inv;
        invZ_o[row] = 1.0f / red[0];
    }
}

// ---------------------------------------------------------------------------
// Pass 4: out[b,h,w] = gamma * invZ[h] * sum_g exp((E[h,g]-mn[h])*inv[h]) * x[b,g,w]
//                      + x[b,h,w]
// WMMA f32 16x16x4 again; A built on the fly (global E + v_exp_f32), B = x
// chunk staged transposed in LDS ([w][k], stride 132 -> conflict-free).
// Block: 8 waves, wave = 16 M-rows x full 128 N-cols (8 tiles).
// ---------------------------------------------------------------------------
__global__ void __launch_bounds__(256) out_kernel(
    const float* __restrict__ x, const float* __restrict__ E,
    const float* __restrict__ mn_, const float* __restrict__ inv_,
    const float* __restrict__ invZ_, const float* __restrict__ gptr,
    float* __restrict__ out) {
    extern __shared__ float ldsT[];   // [128 w][stride 132 k]

    const int hb = blockIdx.x, b = blockIdx.y;
    const int t = threadIdx.x, lane = t & 31, wv = t >> 5;
    const int mlo = lane & 15, half = lane >> 4;

    const size_t rowbase = (size_t)b * HH;
    const int hrowA = hb * 128 + wv * 16 + mlo;     // A-operand row for this lane
    const float mnv = mn_[rowbase + hrowA];
    const float invv = inv_[rowbase + hrowA];
    const float* Erow = E + (rowbase + hrowA) * HH + half * 2;

    v8f acc[8] = {};

    for (int kc = 0; kc < HH; kc += 128) {
        __syncthreads();
        // Stage x[b, kc..kc+128, :] transposed into LDS
        {
            int row = t >> 1, hf = t & 1;
            const float4* src =
                (const float4*)(x + (rowbase + kc + row) * WW + hf * 64);
#pragma unroll
            for (int i = 0; i < 16; ++i) {
                float4 vv = src[i];
                int w0 = hf * 64 + i * 4;
                ldsT[(w0 + 0) * 132 + row] = vv.x;
                ldsT[(w0 + 1) * 132 + row] = vv.y;
                ldsT[(w0 + 2) * 132 + row] = vv.z;
                ldsT[(w0 + 3) * 132 + row] = vv.w;
            }
        }
        __syncthreads();

#pragma unroll 4
        for (int kk = 0; kk < 128; kk += 4) {
            v2f a = *(const v2f*)(Erow + kc + kk);
            a.x = __expf((a.x - mnv) * invv);
            a.y = __expf((a.y - mnv) * invv);
            const float* pB = ldsT + mlo * 132 + kk + half * 2;
#pragma unroll
            for (int nt = 0; nt < 8; ++nt) {
                v2f bv = *(const v2f*)(pB + nt * 16 * 132);
                acc[nt] = __builtin_amdgcn_wmma_f32_16x16x4_f32(
                    false, a, false, bv, (short)0, acc[nt], false, false);
            }
        }
    }

    const float gamma = gptr[0];
#pragma unroll
    for (int nt = 0; nt < 8; ++nt) {
        int col = nt * 16 + mlo;
#pragma unroll
        for (int r = 0; r < 8; ++r) {
            int hrow = hb * 128 + wv * 16 + half * 8 + r;
            float scale = gamma * invZ_[rowbase + hrow];
            float xv = x[(rowbase + hrow) * WW + col];
            out[(rowbase + hrow) * WW + col] = scale * acc[nt][r] + xv;
        }
    }
}

// ---------------------------------------------------------------------------
extern "C" void kernel_launch(void* const* d_in, const int* in_sizes, int n_in,
                              void* d_out, int out_size, void* d_ws, size_t ws_size,
                              hipStream_t stream) {
    const float* x  = (const float*)d_in[0];
    const float* w1 = (const float*)d_in[1];
    const float* b1 = (const float*)d_in[2];
    const float* w2 = (const float*)d_in[3];
    const float* b2 = (const float*)d_in[4];
    const float* g  = (const float*)d_in[5];
    float* out = (float*)d_out;

    // Workspace layout: E (B*H*H f32) | s | mn | inv | invZ   (~269 MB)
    float* E    = (float*)d_ws;
    float* s    = E + (size_t)BB * HH * HH;
    float* mn   = s + (size_t)BB * HH;
    float* inv  = mn + (size_t)BB * HH;
    float* invZ = inv + (size_t)BB * HH;

    rowsum_kernel<<<dim3((BB * HH) / 256), 256, 0, stream>>>(x, s);

    energy_kernel<<<dim3(HH / 128, HH / 128, BB), 256, 2 * 128 * 132 * 4, stream>>>(
        x, s, w1, b1, w2, b2, E);

    stats_kernel<<<dim3(BB * HH), 256, 0, stream>>>(E, mn, inv, invZ);

    out_kernel<<<dim3(HH / 128, BB), 256, 128 * 132 * 4, stream>>>(
        x, E, mn, inv, invZ, g, out);
}